// Restriction_65240553226269
// MI455X (gfx1250) — compile-verified
//
#include <hip/hip_runtime.h>

#define CHUNK 256   // nnz staged per block
#define FDIM  128   // features per row (128 f32 = 512B per row)

__global__ __launch_bounds__(256) void coo_scatter_kernel(
    const float* __restrict__ x_fine,
    const float* __restrict__ r_vals,
    const int*   __restrict__ r_rows,
    const int*   __restrict__ r_cols,
    float*       __restrict__ out,
    int nnz)
{
    __shared__ float s_vals[CHUNK];
    __shared__ int   s_rows[CHUNK];
    __shared__ int   s_cols[CHUNK];

    const int tid   = threadIdx.x;
    const int base  = blockIdx.x * CHUNK;
    const int count = min(CHUNK, nnz - base);

    // ---- Stage COO triples into LDS via gfx1250 async global->LDS copies ----
    // LDS address comes from the VDST VGPR; global addr in 64-bit VGPR pair (GV mode).
    if (tid < count) {
        unsigned lds_v = (unsigned)(unsigned long long)(&s_vals[tid]);
        unsigned lds_r = (unsigned)(unsigned long long)(&s_rows[tid]);
        unsigned lds_c = (unsigned)(unsigned long long)(&s_cols[tid]);
        unsigned long long ga_v = (unsigned long long)(const void*)(r_vals + base + tid);
        unsigned long long ga_r = (unsigned long long)(const void*)(r_rows + base + tid);
        unsigned long long ga_c = (unsigned long long)(const void*)(r_cols + base + tid);
        asm volatile("global_load_async_to_lds_b32 %0, %1, off"
                     :: "v"(lds_v), "v"(ga_v) : "memory");
        asm volatile("global_load_async_to_lds_b32 %0, %1, off"
                     :: "v"(lds_r), "v"(ga_r) : "memory");
        asm volatile("global_load_async_to_lds_b32 %0, %1, off"
                     :: "v"(lds_c), "v"(ga_c) : "memory");
    }
    // Drain this wave's ASYNCcnt, then make LDS visible workgroup-wide.
    asm volatile("s_wait_asynccnt 0x0" ::: "memory");
    __syncthreads();

    // ---- Gather + weighted atomic segment-sum scatter ----
    const int lane = tid & 31;   // wave32
    const int wave = tid >> 5;   // 8 waves per block

    int j = wave;
    // Unroll by 2: two independent 512B row gathers in flight per wave
    // before the atomic burst.
    for (; j + 8 < count; j += 16) {
        const int   row0 = s_rows[j];
        const int   col0 = s_cols[j];
        const float val0 = s_vals[j];
        const int   row1 = s_rows[j + 8];
        const int   col1 = s_cols[j + 8];
        const float val1 = s_vals[j + 8];

        const float4 xv0 = *(const float4*)(x_fine + (size_t)col0 * FDIM + lane * 4);
        const float4 xv1 = *(const float4*)(x_fine + (size_t)col1 * FDIM + lane * 4);

        float* dst0 = out + (size_t)row0 * FDIM + lane * 4;
        float* dst1 = out + (size_t)row1 * FDIM + lane * 4;

        unsafeAtomicAdd(dst0 + 0, val0 * xv0.x);
        unsafeAtomicAdd(dst0 + 1, val0 * xv0.y);
        unsafeAtomicAdd(dst0 + 2, val0 * xv0.z);
        unsafeAtomicAdd(dst0 + 3, val0 * xv0.w);
        unsafeAtomicAdd(dst1 + 0, val1 * xv1.x);
        unsafeAtomicAdd(dst1 + 1, val1 * xv1.y);
        unsafeAtomicAdd(dst1 + 2, val1 * xv1.z);
        unsafeAtomicAdd(dst1 + 3, val1 * xv1.w);
    }
    for (; j < count; j += 8) {
        const int   row = s_rows[j];
        const int   col = s_cols[j];
        const float val = s_vals[j];

        const float4 xv = *(const float4*)(x_fine + (size_t)col * FDIM + lane * 4);
        float* dst = out + (size_t)row * FDIM + lane * 4;

        unsafeAtomicAdd(dst + 0, val * xv.x);
        unsafeAtomicAdd(dst + 1, val * xv.y);
        unsafeAtomicAdd(dst + 2, val * xv.z);
        unsafeAtomicAdd(dst + 3, val * xv.w);
    }
}

__global__ __launch_bounds__(256) void zero_kernel(float* __restrict__ out, int n) {
    int i4 = (blockIdx.x * blockDim.x + threadIdx.x) * 4;
    if (i4 + 3 < n) {
        *(float4*)(out + i4) = make_float4(0.f, 0.f, 0.f, 0.f);
    } else {
        for (int k = i4; k < n; ++k) out[k] = 0.f;
    }
}

extern "C" void kernel_launch(void* const* d_in, const int* in_sizes, int n_in,
                              void* d_out, int out_size, void* d_ws, size_t ws_size,
                              hipStream_t stream) {
    const float* x_fine = (const float*)d_in[0];
    const float* r_vals = (const float*)d_in[1];
    const int*   r_rows = (const int*)d_in[2];
    const int*   r_cols = (const int*)d_in[3];
    float*       out    = (float*)d_out;

    const int nnz = in_sizes[1];

    // 1) Zero the (poisoned) output accumulator.
    const int n4 = (out_size + 3) / 4;
    zero_kernel<<<(n4 + 255) / 256, 256, 0, stream>>>(out, out_size);

    // 2) Async-staged gather + atomic scatter.
    const int blocks = (nnz + CHUNK - 1) / CHUNK;
    coo_scatter_kernel<<<blocks, 256, 0, stream>>>(x_fine, r_vals, r_rows, r_cols, out, nnz);
}